// ForceFieldModel_85143431675990
// MI455X (gfx1250) — compile-verified
//
#include <hip/hip_runtime.h>
#include <math.h>

#define NPAIRS 10
#define NELEM  4

// ---------------------------------------------------------------------------
// Non-returning relaxed agent-scope float atomic add -> global_atomic_add_f32
// (result unused => backend should pick the non-returning form).
// ---------------------------------------------------------------------------
__device__ __forceinline__ void atomic_add_agent(float* p, float v) {
    __hip_atomic_fetch_add(p, v, __ATOMIC_RELAXED, __HIP_MEMORY_SCOPE_AGENT);
}

// ---------------------------------------------------------------------------
// Kernel: per-edge force + scatter-add.  (Defined FIRST so its asm is visible
// in the disassembly snippet.)
//   potential_sf = (A/rho)*exp(-d/rho) - 6*C*(1/d^2)^3 / d
//   coulomb_sf   = q[se]*q[te] * (1/d^2)
//   forces[src] += -(potential_sf + coulomb_sf) * unit_vec        (atomic)
// Parameter tables are staged into LDS via CDNA5 async global->LDS loads.
// ---------------------------------------------------------------------------
__global__ __launch_bounds__(256) void edge_forces(
    const float* __restrict__ raw_charges,  // [3]  Cd, S, Zn (Se = -Cd)
    const float* __restrict__ A,            // [10]
    const float* __restrict__ rho,          // [10]
    const float* __restrict__ C,            // [10]
    const float* __restrict__ dist,         // [E]
    const float* __restrict__ inv_d2,       // [E]
    const float* __restrict__ uv,           // [E,3]
    const int*   __restrict__ edge_src,     // [E]  (edge_index row 0)
    const int*   __restrict__ pair_idx,     // [E]
    const int*   __restrict__ src_el,       // [E]
    const int*   __restrict__ tgt_el,       // [E]
    float*       __restrict__ out,          // [n_nodes,3]
    int n_edges)
{
    // Raw staging (filled by async DMA), then derived tables.
    __shared__ float sA[NPAIRS], sRho[NPAIRS], sC[NPAIRS], sRC[4];
    __shared__ float sAr[NPAIRS];      // A/rho        (same division as reference)
    __shared__ float sNegRinv[NPAIRS]; // -1/rho       (exp argument scale)
    __shared__ float sC6[NPAIRS];      // 6*C
    __shared__ float sQQ[NELEM * NELEM]; // q[i]*q[j]

    const int t = threadIdx.x;

#if __has_builtin(__builtin_amdgcn_global_load_async_to_lds_b32)
    // CDNA5 async global->LDS path (ASYNCcnt-tracked, no VGPR data movement).
    typedef __attribute__((address_space(1))) int* g32_t;
    typedef __attribute__((address_space(3))) int* l32_t;
    if (t < NPAIRS) {
        __builtin_amdgcn_global_load_async_to_lds_b32((g32_t)(A   + t), (l32_t)(sA   + t), 0, 0);
        __builtin_amdgcn_global_load_async_to_lds_b32((g32_t)(rho + t), (l32_t)(sRho + t), 0, 0);
        __builtin_amdgcn_global_load_async_to_lds_b32((g32_t)(C   + t), (l32_t)(sC   + t), 0, 0);
    }
    if (t < 3) {
        __builtin_amdgcn_global_load_async_to_lds_b32((g32_t)(raw_charges + t), (l32_t)(sRC + t), 0, 0);
    }
#if __has_builtin(__builtin_amdgcn_s_wait_asynccnt)
    __builtin_amdgcn_s_wait_asynccnt(0);
#endif
#else
    // Fallback: plain loads.
    if (t < NPAIRS) { sA[t] = A[t]; sRho[t] = rho[t]; sC[t] = C[t]; }
    if (t < 3)      { sRC[t] = raw_charges[t]; }
#endif
    __syncthreads();

    // Derive per-pair constants and the 4x4 charge-product table.
    if (t < NPAIRS) {
        const float r = sRho[t];
        sAr[t]      = sA[t] / r;     // identical rounding to reference's A_p/rho_p
        sNegRinv[t] = -1.0f / r;
        sC6[t]      = 6.0f * sC[t];
    }
    if (t < NELEM * NELEM) {
        const float rc0 = sRC[0];
        float q[NELEM];
        q[0] = rc0;      // Cd
        q[1] = -rc0;     // Se = -Cd
        q[2] = sRC[1];   // S
        q[3] = sRC[2];   // Zn
        sQQ[t] = q[t >> 2] * q[t & 3];
    }
    __syncthreads();

    const int e = blockIdx.x * blockDim.x + t;
    if (e >= n_edges) return;

    // Coalesced edge stream (memory-bound: ~36 B/edge).
    const float d  = dist[e];
    const float i2 = inv_d2[e];
    const int   p  = pair_idx[e];
    const int   se = src_el[e];
    const int   te = tgt_el[e];
    const int   s  = edge_src[e];

    const float i6   = i2 * i2 * i2;
    const float pot  = sAr[p] * expf(d * sNegRinv[p]) - sC6[p] * i6 / d;
    const float coul = sQQ[(se << 2) | te] * i2;
    const float f    = -(pot + coul);

    const float ux = uv[3 * e + 0];
    const float uy = uv[3 * e + 1];
    const float uz = uv[3 * e + 2];

    float* o = out + 3 * s;
    atomic_add_agent(o + 0, f * ux);
    atomic_add_agent(o + 1, f * uy);
    atomic_add_agent(o + 2, f * uz);
}

// ---------------------------------------------------------------------------
// Kernel: zero the force accumulator (d_out is poisoned by the harness).
// ---------------------------------------------------------------------------
__global__ __launch_bounds__(256) void zero_f32(float* __restrict__ out, int n) {
    int i = blockIdx.x * blockDim.x + threadIdx.x;
    if (i < n) out[i] = 0.0f;
}

// ---------------------------------------------------------------------------
// Host launcher.
// Input order (setup_inputs dict):
//  0 raw_charges[3] f32, 1 A[10] f32, 2 rho[10] f32, 3 C[10] f32,
//  4 distances[E] f32, 5 inverse_distances_sq[E] f32, 6 edge_unit_vectors[E*3] f32,
//  7 edge_index[2*E] i32, 8 pair_indices[E] i32, 9 source_elements[E] i32,
//  10 target_elements[E] i32, 11 n_nodes (scalar)
// Output: forces [n_nodes*3] f32.
// ---------------------------------------------------------------------------
extern "C" void kernel_launch(void* const* d_in, const int* in_sizes, int n_in,
                              void* d_out, int out_size, void* d_ws, size_t ws_size,
                              hipStream_t stream) {
    const float* raw_charges = (const float*)d_in[0];
    const float* A           = (const float*)d_in[1];
    const float* rho         = (const float*)d_in[2];
    const float* C           = (const float*)d_in[3];
    const float* dist        = (const float*)d_in[4];
    const float* inv_d2      = (const float*)d_in[5];
    const float* uv          = (const float*)d_in[6];
    const int*   edge_index  = (const int*)d_in[7];   // row 0 = first E entries
    const int*   pair_idx    = (const int*)d_in[8];
    const int*   src_el      = (const int*)d_in[9];
    const int*   tgt_el      = (const int*)d_in[10];
    float*       out         = (float*)d_out;

    const int n_edges = in_sizes[4];

    zero_f32<<<(out_size + 255) / 256, 256, 0, stream>>>(out, out_size);

    edge_forces<<<(n_edges + 255) / 256, 256, 0, stream>>>(
        raw_charges, A, rho, C, dist, inv_d2, uv,
        edge_index /* row 0 */, pair_idx, src_el, tgt_el,
        out, n_edges);
}